// CrossAttention_79164837200441
// MI455X (gfx1250) — compile-verified
//
#include <hip/hip_runtime.h>

typedef float v2f __attribute__((ext_vector_type(2)));
typedef float v8f __attribute__((ext_vector_type(8)));

#define C_DIM 128
#define GRP 8
#define NSAMP 16
#define CSH 16

// out[n][c] = sum_k in[n][k] * W[c][k] + bias[c]
// One block = one 16-row tile; wave w (0..7) computes output columns [16w,16w+16).
// K accumulated via 32 chained V_WMMA_F32_16X16X4_F32.
__global__ __launch_bounds__(256) void qkv_gemm_kernel(
    const float* __restrict__ in, const float* __restrict__ W,
    const float* __restrict__ bias, float* __restrict__ out, int nrows)
{
  const int lane    = threadIdx.x & 31;
  const int colTile = threadIdx.x >> 5;     // 0..7 -> output col tile
  const int row0    = blockIdx.x * 16;

  const int m  = lane & 15;                 // M (A) / N (B,D) index
  const int kh = (lane >> 4) << 1;          // K sub-offset: 0 (lanes 0-15), 2 (lanes 16-31)

  int arow = row0 + m;
  if (arow >= nrows) arow = nrows - 1;      // clamp (keeps EXEC all-ones for WMMA)
  const float* __restrict__ aptr = in + (size_t)arow * C_DIM;
  const float* __restrict__ bptr = W  + (size_t)(colTile * 16 + m) * C_DIM;

  v8f acc = {};
#pragma unroll
  for (int kk = 0; kk < C_DIM; kk += 4) {
    // A 16x4 layout: VGPR0 = K=kh, VGPR1 = K=kh+1 (per half-wave)
    float2 af = *(const float2*)(aptr + kk + kh);
    // B 4x16 layout (symmetric): lane = N, VGPR0/1 = rows K=kh, kh+1; B[k][n] = W[n][k]
    float2 bf = *(const float2*)(bptr + kk + kh);
    v2f a; a[0] = af.x; a[1] = af.y;
    v2f b; b[0] = bf.x; b[1] = bf.y;
    acc = __builtin_amdgcn_wmma_f32_16x16x4_f32(false, a, false, b,
                                                (short)0, acc, false, false);
  }

  // D layout: VGPR v holds M = (lane>>4)*8 + v, N = lane&15
  const int   dn = colTile * 16 + m;
  const float bv = bias[dn];
  const int   mb = (lane >> 4) << 3;
#pragma unroll
  for (int v = 0; v < 8; ++v) {
    int drow = row0 + mb + v;
    if (drow < nrows)
      out[(size_t)drow * C_DIM + dn] = acc[v] + bv;
  }
}

// One wave per query point. lane -> (group g = lane>>2, 4 channels of that group).
// Pass 1: gather K, positional MLP, grouped logits (reduced across 4 lanes).
// Register softmax over S=16, then pass 2 gathers V (hits L2) and accumulates.
__global__ __launch_bounds__(256) void attn_kernel(
    const float* __restrict__ p, const float* __restrict__ p2,
    const int* __restrict__ idx,
    const float* __restrict__ Qf, const float* __restrict__ Kf,
    const float* __restrict__ Vf,
    const float* __restrict__ W1, const float* __restrict__ b1,
    const float* __restrict__ bng, const float* __restrict__ bnb,
    const float* __restrict__ bnm, const float* __restrict__ bnv,
    const float* __restrict__ W2, const float* __restrict__ b2,
    float* __restrict__ out, int n)
{
  const int lane = threadIdx.x & 31;
  const int pt   = blockIdx.x * 8 + (threadIdx.x >> 5);
  if (pt >= n) return;

  const int g  = lane >> 2;                 // group 0..7
  const int cb = g * CSH + (lane & 3) * 4;  // first of this lane's 4 channels

  float4 q = *(const float4*)(Qf + (size_t)pt * C_DIM + cb);

  // linear_p parameters (per-lane broadcast copies; BN folded into scale/shift)
  float w1[9], bb1[3], sc[3], bt[3], mn[3];
#pragma unroll
  for (int i = 0; i < 9; ++i) w1[i] = W1[i];
#pragma unroll
  for (int i = 0; i < 3; ++i) {
    bb1[i] = b1[i];
    sc[i]  = bng[i] * rsqrtf(bnv[i] + 1e-5f);
    bt[i]  = bnb[i];
    mn[i]  = bnm[i];
  }
  float w2q[3], w2k[3];
#pragma unroll
  for (int i = 0; i < 3; ++i) {
    w2q[i] = W2[g * 3 + i];            // p_rq row for this group
    w2k[i] = W2[(GRP + g) * 3 + i];    // p_rk row for this group
  }
  const float b2q = b2[g], b2k = b2[GRP + g];

  const float px = p[pt*3+0], py = p[pt*3+1], pz = p[pt*3+2];

  int   js[NSAMP];
  float lg[NSAMP];
#pragma unroll
  for (int s = 0; s < NSAMP; ++s) {
    const int j = idx[pt*NSAMP + s];
    js[s] = j;
    const float rx = p2[j*3+0] - px;
    const float ry = p2[j*3+1] - py;
    const float rz = p2[j*3+2] - pz;
    float h0 = (w1[0]*rx + w1[1]*ry + w1[2]*rz + bb1[0] - mn[0]) * sc[0] + bt[0];
    float h1 = (w1[3]*rx + w1[4]*ry + w1[5]*rz + bb1[1] - mn[1]) * sc[1] + bt[1];
    float h2 = (w1[6]*rx + w1[7]*ry + w1[8]*rz + bb1[2] - mn[2]) * sc[2] + bt[2];
    h0 = fmaxf(h0, 0.0f); h1 = fmaxf(h1, 0.0f); h2 = fmaxf(h2, 0.0f);
    const float rq = w2q[0]*h0 + w2q[1]*h1 + w2q[2]*h2 + b2q;
    const float rk = w2k[0]*h0 + w2k[1]*h1 + w2k[2]*h2 + b2k;

    float4 kv = *(const float4*)(Kf + (size_t)j * C_DIM + cb);
    float part = (kv.x + rk) * (q.x + rq) + (kv.y + rk) * (q.y + rq)
               + (kv.z + rk) * (q.z + rq) + (kv.w + rk) * (q.w + rq);
    // reduce partial dot across the 4 lanes of this group
    part += __shfl_xor(part, 1, 4);
    part += __shfl_xor(part, 2, 4);
    lg[s] = part * 0.25f;               // 1/sqrt(CS) = 1/4
  }

  // softmax over neighbors (in registers; all 4 lanes of a group agree)
  float mx = lg[0];
#pragma unroll
  for (int s = 1; s < NSAMP; ++s) mx = fmaxf(mx, lg[s]);
  float sum = 0.0f;
#pragma unroll
  for (int s = 0; s < NSAMP; ++s) { lg[s] = __expf(lg[s] - mx); sum += lg[s]; }
  const float inv = 1.0f / sum;

  float4 o = make_float4(0.0f, 0.0f, 0.0f, 0.0f);
#pragma unroll
  for (int s = 0; s < NSAMP; ++s) {
    const float w = lg[s] * inv;
    float4 vv = *(const float4*)(Vf + (size_t)js[s] * C_DIM + cb);
    o.x += w * vv.x; o.y += w * vv.y; o.z += w * vv.z; o.w += w * vv.w;
  }
  *(float4*)(out + (size_t)pt * C_DIM + cb) = o;
}

extern "C" void kernel_launch(void* const* d_in, const int* in_sizes, int n_in,
                              void* d_out, int out_size, void* d_ws, size_t ws_size,
                              hipStream_t stream) {
  const float* p   = (const float*)d_in[0];
  const float* x   = (const float*)d_in[1];
  const float* p2  = (const float*)d_in[2];
  const float* x2  = (const float*)d_in[3];
  const int*   idx = (const int*)  d_in[4];
  const float* Wq  = (const float*)d_in[5];
  const float* bq  = (const float*)d_in[6];
  const float* Wk  = (const float*)d_in[7];
  const float* bk  = (const float*)d_in[8];
  const float* Wv  = (const float*)d_in[9];
  const float* bv  = (const float*)d_in[10];
  const float* W1  = (const float*)d_in[11];
  const float* b1  = (const float*)d_in[12];
  const float* bng = (const float*)d_in[13];
  const float* bnb = (const float*)d_in[14];
  const float* bnm = (const float*)d_in[15];
  const float* bnv = (const float*)d_in[16];
  const float* W2  = (const float*)d_in[17];
  const float* b2  = (const float*)d_in[18];
  float* out = (float*)d_out;

  const int n  = in_sizes[0] / 3;   // query points
  const int n2 = in_sizes[2] / 3;   // key/value points

  float* Qw = (float*)d_ws;
  float* Kw = Qw + (size_t)n  * C_DIM;
  float* Vw = Kw + (size_t)n2 * C_DIM;

  dim3 blk(256);
  qkv_gemm_kernel<<<dim3((n  + 15) / 16), blk, 0, stream>>>(x,  Wq, bq, Qw, n);
  qkv_gemm_kernel<<<dim3((n2 + 15) / 16), blk, 0, stream>>>(x2, Wk, bk, Kw, n2);
  qkv_gemm_kernel<<<dim3((n2 + 15) / 16), blk, 0, stream>>>(x2, Wv, bv, Vw, n2);
  attn_kernel<<<dim3((n + 7) / 8), blk, 0, stream>>>(
      p, p2, idx, Qw, Kw, Vw, W1, b1, bng, bnb, bnm, bnv, W2, b2, out, n);
}